// LSTM_10428180595353
// MI455X (gfx1250) — compile-verified
//
#include <hip/hip_runtime.h>

// ---------------------------------------------------------------------------
// LSTM persistent kernel for MI455X (gfx1250).
//   B=64, T=1024, IN=256, HID=256.  out[t] == h_t  (float32, [T,B,HID])
//
// Design:
//  * 16 workgroups x 128 threads (4 waves). Wave (mrow=waveId, jtile=blockIdx)
//    owns output block rows b in [mrow*16,+16), cols j in [jtile*16,+16) for
//    ALL 4 gates -> LSTM cell math fully in registers, c-state in VGPRs.
//  * Gate weights (4 x 512 x 16 slice) staged once into LDS (128 KB) in
//    V_WMMA_F32_16X16X4_F32 B-fragment order, gate-major [g][kb][lane].
//    Four opaque *address-space(3)* per-lane gate base registers keep every
//    weight load a ds_load_b64 with a 16-bit immediate offset.
//  * out[t] doubles as h_t: recurrent A-fragments read straight from out.
//  * Split grid barrier (arrive/wait): the h-independent x-projection GEMM of
//    step t+1 executes between arrive and wait, hiding barrier latency.
// ---------------------------------------------------------------------------

#define B_    64
#define T_    1024
#define IN_   256
#define HID_  256
#define BH    (B_ * HID_)
#define NWG   16
#define NTHR  128   // 4 waves

typedef float v2f __attribute__((ext_vector_type(2)));
typedef float v8f __attribute__((ext_vector_type(8)));

// 32-bit LDS (address space 3) pointer to a primitive vector type: loads
// through it lower to DS ops and copy cleanly out of AS(3).
typedef __attribute__((address_space(3))) const v2f* lds_cv2p;
typedef __attribute__((address_space(3))) v2f*       lds_v2p;

__device__ __forceinline__ float fast_sigmoid(float x) {
    return 1.0f / (1.0f + __expf(-x));
}

__device__ __forceinline__ float fast_tanh(float x) {
    x = fminf(fmaxf(x, -15.0f), 15.0f);
    float e = __expf(2.0f * x);
    return (e - 1.0f) / (e + 1.0f);
}

// Opaque register barrier on a 32-bit LDS pointer: stops LLVM from folding the
// four gate bases into one base + >16-bit constants (per-load v_add's), while
// keeping the access on the DS path (AS3, not flat).
__device__ __forceinline__ lds_cv2p opaque_lds(lds_cv2p p) {
    asm("" : "+v"(p));
    return p;
}

// Signal arrival at the step barrier. Returns the generation observed before
// arriving (meaningful in thread 0 only).
__device__ __forceinline__ unsigned barrier_arrive(unsigned* cnt, unsigned* gen) {
    __threadfence();          // make this step's h stores device-visible
    __syncthreads();
    unsigned g0 = 0;
    if (threadIdx.x == 0) {
        g0 = __hip_atomic_load(gen, __ATOMIC_ACQUIRE, __HIP_MEMORY_SCOPE_AGENT);
        unsigned arrived = __hip_atomic_fetch_add(cnt, 1u, __ATOMIC_ACQ_REL,
                                                  __HIP_MEMORY_SCOPE_AGENT);
        if (arrived == NWG - 1) {
            __hip_atomic_store(cnt, 0u, __ATOMIC_RELAXED, __HIP_MEMORY_SCOPE_AGENT);
            __hip_atomic_fetch_add(gen, 1u, __ATOMIC_ACQ_REL, __HIP_MEMORY_SCOPE_AGENT);
        }
    }
    return g0;
}

__device__ __forceinline__ void barrier_wait(unsigned* gen, unsigned g0) {
    if (threadIdx.x == 0) {
        while (__hip_atomic_load(gen, __ATOMIC_ACQUIRE,
                                 __HIP_MEMORY_SCOPE_AGENT) == g0) {
            __builtin_amdgcn_s_sleep(1);
        }
    }
    __syncthreads();
}

extern "C" __global__ void lstm_bar_init(unsigned* bar) {
    bar[0] = 0u;
    bar[1] = 0u;
}

// One K-block (4 k-values) of the 4-gate GEMM: acc_g += A_frag x B_frag(g).
// wgN are opaque per-lane AS3 gate bases; fragment for k-block kb sits at
// element offset kb*32 (byte offset kb*256 <= 32512: 16-bit DS immediate).
__device__ __forceinline__ void gemm4_step(v8f& a0, v8f& a1, v8f& a2, v8f& a3,
                                           v2f a,
                                           lds_cv2p wg0, lds_cv2p wg1,
                                           lds_cv2p wg2, lds_cv2p wg3,
                                           int kb) {
    v2f b0 = wg0[kb * 32];
    v2f b1 = wg1[kb * 32];
    v2f b2 = wg2[kb * 32];
    v2f b3 = wg3[kb * 32];
    a0 = __builtin_amdgcn_wmma_f32_16x16x4_f32(false, a, false, b0, (short)0, a0, false, false);
    a1 = __builtin_amdgcn_wmma_f32_16x16x4_f32(false, a, false, b1, (short)0, a1, false, false);
    a2 = __builtin_amdgcn_wmma_f32_16x16x4_f32(false, a, false, b2, (short)0, a2, false, false);
    a3 = __builtin_amdgcn_wmma_f32_16x16x4_f32(false, a, false, b3, (short)0, a3, false, false);
}

extern "C" __global__ __launch_bounds__(NTHR)
void lstm_persistent(const float* __restrict__ x,
                     const float* __restrict__ Wi, const float* __restrict__ bi,
                     const float* __restrict__ Wf, const float* __restrict__ bf,
                     const float* __restrict__ Wo, const float* __restrict__ bo,
                     const float* __restrict__ Wc, const float* __restrict__ bc,
                     float* __restrict__ out,
                     unsigned* __restrict__ bar) {
    // [4 gates][128 kblocks][32 lanes] v2f = 128 KB
    extern __shared__ v2f ldsW[];

    const int tid   = threadIdx.x;
    const int lane  = tid & 31;
    const int mrow  = tid >> 5;        // 0..3  (batch row block)
    const int jtile = blockIdx.x;      // 0..15 (hidden col block)
    const int hh    = lane >> 4;       // lane half: selects K/M sub-group
    const int l15   = lane & 15;
    const int jcol  = jtile * 16 + l15;

    // ---- Stage this WG's weight slice into LDS in B-fragment order --------
    // Element e = (g*128 + kb)*32 + ln  holds, for lane ln:
    //   ( W_g[kb*4 + 2*(ln>>4)][ncol], W_g[kb*4 + 2*(ln>>4) + 1][ncol] )
    const float* Wg[4] = {Wi, Wf, Wo, Wc};
    for (int e = tid; e < 4 * 128 * 32; e += NTHR) {
        int ln   = e & 31;
        int kb   = (e >> 5) & 127;
        int g    = e >> 12;
        int nh   = ln >> 4;
        int ncol = jtile * 16 + (ln & 15);
        int r0   = kb * 4 + 2 * nh;            // 0..511 (rows of W: [h | x])
        const float* W = Wg[g];
        v2f w;
        w.x = W[(size_t)r0 * HID_ + ncol];
        w.y = W[(size_t)(r0 + 1) * HID_ + ncol];
        ldsW[e] = w;
    }
    __syncthreads();

    // Per-lane bias values for this j column (folded into acc init).
    const float vbi = bi[jcol], vbf = bf[jcol], vbo = bo[jcol], vbc = bc[jcol];

    // A-fragment row (batch index) for this lane.
    const int m = mrow * 16 + l15;
    const float* pxm = x + (size_t)m * T_ * IN_ + 2 * hh;   // x[m, :, :]

    // Opaque per-lane AS3 gate bases. h-part uses kb 0..63 (offsets
    // 0..16128 B), x-part kb 64..127 (16384..32512 B): 16-bit immediates.
    lds_cv2p wg0 = opaque_lds((lds_cv2p)(ldsW + lane + 0 * 128 * 32));
    lds_cv2p wg1 = opaque_lds((lds_cv2p)(ldsW + lane + 1 * 128 * 32));
    lds_cv2p wg2 = opaque_lds((lds_cv2p)(ldsW + lane + 2 * 128 * 32));
    lds_cv2p wg3 = opaque_lds((lds_cv2p)(ldsW + lane + 3 * 128 * 32));

    // Output offset for this lane's first accumulator row.
    const size_t ooff = (size_t)(mrow * 16 + 8 * hh) * HID_ + jcol;

    // Cell state lives in registers: element (v, lane) covers
    // b = mrow*16 + v + 8*hh, j = jcol.
    float cst[8];
#pragma unroll
    for (int v = 0; v < 8; ++v) cst[v] = 0.0f;

    v8f acc0, acc1, acc2, acc3;
#pragma unroll
    for (int v = 0; v < 8; ++v) { acc0[v] = vbi; acc1[v] = vbf; acc2[v] = vbo; acc3[v] = vbc; }

    // Prologue: x-projection for t = 0 (h_{-1} == 0, so this is the full acc).
    {
        const float* pxt = pxm;                 // t = 0
#pragma unroll 8
        for (int kb = 0; kb < 64; ++kb) {
            v2f a = *(const v2f*)(pxt + kb * 4);
            gemm4_step(acc0, acc1, acc2, acc3, a, wg0, wg1, wg2, wg3, 64 + kb);
        }
    }

    unsigned g0 = 0;
    for (int t = 0; t < T_; ++t) {
        // ---- recurrent part: K = 0..255, A = h_{t-1} = out[t-1] ----------
        if (t > 0) {
            barrier_wait(bar + 1, g0);          // h_{t-1} now visible
            const float* ph = out + (size_t)(t - 1) * BH + (size_t)m * HID_ + 2 * hh;
#pragma unroll 8
            for (int kb = 0; kb < 64; ++kb) {
                v2f a = *(const v2f*)(ph + kb * 4);
                gemm4_step(acc0, acc1, acc2, acc3, a, wg0, wg1, wg2, wg3, kb);
            }
        }

        // ---- element-wise LSTM cell (all 4 gates in-wave) ----------------
        float* po = out + (size_t)t * BH + ooff;
#pragma unroll
        for (int v = 0; v < 8; ++v) {
            float gi = fast_sigmoid(acc0[v]);
            float gf = fast_sigmoid(acc1[v]);
            float go = fast_sigmoid(acc2[v]);
            float gc = fast_tanh(acc3[v]);
            float cn = gf * cst[v] + gi * gc;
            cst[v] = cn;
            float hn = go * fast_tanh(cn);
            po[(size_t)v * HID_] = hn;          // C/D layout: M = v + 8*(lane>>4)
        }

        if (t + 1 < T_) {
            // Signal our h_t stores, then hide barrier latency behind the
            // h-independent x-projection GEMM for step t+1.
            g0 = barrier_arrive(bar, bar + 1);

#pragma unroll
            for (int v = 0; v < 8; ++v) {
                acc0[v] = vbi; acc1[v] = vbf; acc2[v] = vbo; acc3[v] = vbc;
            }
            const float* pxt = pxm + (size_t)(t + 1) * IN_;
#pragma unroll 8
            for (int kb = 0; kb < 64; ++kb) {
                v2f a = *(const v2f*)(pxt + kb * 4);
                gemm4_step(acc0, acc1, acc2, acc3, a, wg0, wg1, wg2, wg3, 64 + kb);
            }
        }
    }
}

extern "C" void kernel_launch(void* const* d_in, const int* in_sizes, int n_in,
                              void* d_out, int out_size, void* d_ws, size_t ws_size,
                              hipStream_t stream) {
    const float* x  = (const float*)d_in[0];
    const float* Wi = (const float*)d_in[1];
    const float* bi = (const float*)d_in[2];
    const float* Wf = (const float*)d_in[3];
    const float* bf = (const float*)d_in[4];
    const float* Wo = (const float*)d_in[5];
    const float* bo = (const float*)d_in[6];
    const float* Wc = (const float*)d_in[7];
    const float* bc = (const float*)d_in[8];
    float* out = (float*)d_out;
    unsigned* bar = (unsigned*)d_ws;

    lstm_bar_init<<<1, 1, 0, stream>>>(bar);

    size_t shmem = (size_t)4 * 128 * 32 * sizeof(v2f);   // 128 KB
    lstm_persistent<<<dim3(NWG), dim3(NTHR), shmem, stream>>>(
        x, Wi, bi, Wf, bf, Wo, bo, Wc, bc, out, bar);
}